// MaskCrossScaleAttention_41618233098977
// MI455X (gfx1250) — compile-verified
//
#include <hip/hip_runtime.h>
#include <hip/hip_bf16.h>
#include <math.h>

// ---------------------------------------------------------------------------
// MaskCrossScaleAttention for MI455X (gfx1250, wave32, WMMA + TDM).
// Hot path: fused sigmoid-attention, two 4096x4096x128 GEMMs per batch with
// V_WMMA_F32_16X16X4_F32; K/V tiles staged by the Tensor Data Mover
// (tensor_load_to_lds) into double-buffered LDS, with D# pad producing the
// bank-conflict-free 17-float row stride in hardware.
// ---------------------------------------------------------------------------

typedef __attribute__((ext_vector_type(2))) float v2f;
typedef __attribute__((ext_vector_type(8))) float v8f;
typedef __attribute__((ext_vector_type(4))) unsigned int u32x4;
typedef __attribute__((ext_vector_type(8))) int i32x8;
typedef __attribute__((ext_vector_type(4))) int i32x4;

#define BATCH 4
#define NPIX  4096      // 64*64 attention positions
#define CATT  128       // attention channels
#define EPSV  1e-5f
#define LROW  17        // padded LDS row stride (floats)

// Raw LDS byte offset (addrspace(3)) for TDM descriptors.
__device__ inline unsigned lds_off(const void* p) {
    return (unsigned)(size_t)(__attribute__((address_space(3))) const void*)p;
}

// Issue one TDM load of a 128x16 f32 tile (rows stride NPIX in memory) into
// LDS at lds_addr, padding each 16-DWORD row with 1 DWORD (-> 17-float rows).
// D# packing per CDNA5 ISA.
__device__ inline void tdm_load_tile(unsigned lds_addr, const float* gptr) {
    unsigned long long ga = (unsigned long long)(size_t)gptr;
    u32x4 g0;
    g0[0] = 1u;                                        // count=1, user mode
    g0[1] = lds_addr;                                  // lds_addr[31:0]
    g0[2] = (unsigned)(ga & 0xFFFFFFFFu);              // global_addr[31:0]
    g0[3] = (unsigned)((ga >> 32) & 0x01FFFFFFu)       // global_addr[56:32]
          | (2u << 30);                                // type = 2 ("image")
    i32x8 g1;
    g1[0] = (int)((2u << 16)      // data_size = 4 bytes
          | (1u << 20)            // pad_enable
          | (3u << 22));          // pad_interval: 16 DWORDs  (pad_amount=0 -> 1 DW)
    g1[1] = (int)(4096u << 16);   // tensor_dim0[15:0] = 4096 (cols)
    g1[2] = (int)(128u << 16);    // tensor_dim0 hi = 0; tensor_dim1 = 128 (rows)
    g1[3] = (int)(16u << 16);     // tensor_dim1 hi = 0; tile_dim0 = 16
    g1[4] = 128;                  // tile_dim1 = 128; tile_dim2 = 0
    g1[5] = 4096;                 // tensor_dim0_stride[31:0] = 4096
    g1[6] = 0;                    // stride hi / tensor_dim1_stride lo
    g1[7] = 0;
    i32x4 z4 = {0, 0, 0, 0};      // groups 2/3 unused (2D tensor)
    i32x8 z8 = {0, 0, 0, 0, 0, 0, 0, 0};
    __builtin_amdgcn_tensor_load_to_lds(g0, g1, z4, z4, z8, 0);
}

// ---------------------------------------------------------------------------
// Kernel 1: conv3x3 (64->16, pad 1) + BN + PReLU  -> conv_out (4,16,128,128)
// ---------------------------------------------------------------------------
__global__ __launch_bounds__(256)
void k_conv3x3(const float* __restrict__ em, const float* __restrict__ w,
               const float* __restrict__ bconv,
               const float* __restrict__ g, const float* __restrict__ bb,
               const float* __restrict__ mm, const float* __restrict__ vv,
               const float* __restrict__ prelu1, float* __restrict__ conv_out)
{
    int idx = blockIdx.x * 256 + threadIdx.x;           // 4*16*128*128
    int ow = idx & 127;
    int oh = (idx >> 7) & 127;
    int oc = (idx >> 14) & 15;
    int b  = idx >> 18;

    float sum = 0.0f;
    for (int ic = 0; ic < 64; ++ic) {
        const float* in = em + ((size_t)(b * 64 + ic) << 14);   // 128*128
        const float* wp = w + ((oc * 64 + ic) * 9);
        #pragma unroll
        for (int kh = 0; kh < 3; ++kh) {
            int ih = oh + kh - 1;
            if ((unsigned)ih >= 128u) continue;
            #pragma unroll
            for (int kw = 0; kw < 3; ++kw) {
                int iw = ow + kw - 1;
                if ((unsigned)iw >= 128u) continue;
                sum += in[(ih << 7) + iw] * wp[kh * 3 + kw];
            }
        }
    }
    float scale = g[oc] * rsqrtf(vv[oc] + EPSV);
    float y = (sum + bconv[oc] - mm[oc]) * scale + bb[oc];
    float a = prelu1[0];
    conv_out[idx] = (y >= 0.0f) ? y : a * y;
}

// ---------------------------------------------------------------------------
// Kernel 2: maxpool 3x3 s2 p1 + depthwise scale -> f channels [0,16)
// ---------------------------------------------------------------------------
__global__ __launch_bounds__(256)
void k_pool_dw(const float* __restrict__ conv_out,
               const float* __restrict__ dw_w, const float* __restrict__ dw_b,
               float* __restrict__ f)
{
    int idx = blockIdx.x * 256 + threadIdx.x;           // 4*16*64*64
    int ow = idx & 63;
    int oh = (idx >> 6) & 63;
    int c  = (idx >> 12) & 15;
    int b  = idx >> 16;

    const float* in = conv_out + ((size_t)(b * 16 + c) << 14);
    float mx = -INFINITY;
    #pragma unroll
    for (int kh = 0; kh < 3; ++kh) {
        int ih = 2 * oh + kh - 1;
        if ((unsigned)ih >= 128u) continue;
        #pragma unroll
        for (int kw = 0; kw < 3; ++kw) {
            int iw = 2 * ow + kw - 1;
            if ((unsigned)iw >= 128u) continue;
            mx = fmaxf(mx, in[(ih << 7) + iw]);
        }
    }
    f[((size_t)(b * 48 + c) << 12) + (oh << 6) + ow] = mx * dw_w[c] + dw_b[c];
}

// ---------------------------------------------------------------------------
// Kernel 3: de_forward 1x1 conv 128->32 -> f channels [16,48)
// ---------------------------------------------------------------------------
__global__ __launch_bounds__(256)
void k_de(const float* __restrict__ dm, const float* __restrict__ de_w,
          const float* __restrict__ de_b, float* __restrict__ f)
{
    int idx = blockIdx.x * 256 + threadIdx.x;           // 4*32*4096
    int i  = idx & 4095;
    int oc = (idx >> 12) & 31;
    int b  = idx >> 17;

    float sum = de_b[oc];
    const float* wp = de_w + oc * 128;
    const float* in = dm + ((size_t)b << 19);           // 128*4096
    for (int ic = 0; ic < 128; ++ic)
        sum += wp[ic] * in[((size_t)ic << 12) + i];
    f[((size_t)(b * 48 + 16 + oc) << 12) + i] = sum;
}

// ---------------------------------------------------------------------------
// Kernel 4: QKV 1x1 projections 48->128 with GEMM-friendly layouts:
//   qT : [b][i][c]   (i-major; A-fragment loads contiguous)
//   kM : [b][c][j]
//   vM : [b][c][j]
// ---------------------------------------------------------------------------
__global__ __launch_bounds__(256)
void k_qkv(const float* __restrict__ f,
           const float* __restrict__ q_w, const float* __restrict__ q_b,
           const float* __restrict__ k_w, const float* __restrict__ k_b,
           const float* __restrict__ v_w, const float* __restrict__ v_b,
           float* __restrict__ qT, float* __restrict__ kM, float* __restrict__ vM)
{
    int idx = blockIdx.x * 256 + threadIdx.x;           // 4*4096*128
    int c = idx & 127;
    int i = (idx >> 7) & 4095;
    int b = idx >> 19;

    float aq = q_b[c], ak = k_b[c], av = v_b[c];
    const float* fb = f + ((size_t)b * 48 << 12) + i;
    #pragma unroll 4
    for (int ci = 0; ci < 48; ++ci) {
        float x = fb[(size_t)ci << 12];
        aq += q_w[c * 48 + ci] * x;
        ak += k_w[c * 48 + ci] * x;
        av += v_w[c * 48 + ci] * x;
    }
    qT[(((size_t)b * NPIX) + i) * CATT + c] = aq;
    kM[(((size_t)b * CATT) + c) * NPIX + i] = ak;
    vM[(((size_t)b * CATT) + c) * NPIX + i] = av;
}

// ---------------------------------------------------------------------------
// Kernel 5: fused sigmoid-attention (WMMA + TDM double buffering).
//   S[i,j]   = sum_c qT[i,c]*kM[c,j]          (32 x wmma per 16x16 j-tile)
//   P        = sigmoid(S)                     (elementwise on D frags)
//   O[i,c]  += sum_j P[i,j]*vM[c,j]           (8x4 x wmma per j-tile)
// 8 waves/block, each wave owns 16 rows of i. Wave 0 drives the TDM:
// while tile j is consumed, tile j+16 streams into the other LDS buffer.
// Fragment layouts per CDNA5 ISA 7.12.2:
//   A 16x4 : VGPR = K%2, half = K/2, lane-in-half = M
//   B 4x16 : VGPR = K%2, half = K/2, lane-in-half = N
//   C/D    : VGPR = M%8, half = M/8, lane-in-half = N
// ---------------------------------------------------------------------------
__global__ __launch_bounds__(256)
void k_attention(const float* __restrict__ qT, const float* __restrict__ kM,
                 const float* __restrict__ vM, float* __restrict__ attn)
{
    __shared__ float sK[2][CATT * LROW];   // K tile [c][j], TDM-padded rows
    __shared__ float sV[2][CATT * LROW];   // V tile [c][j], TDM-padded rows
    __shared__ float sS[8 * 16 * LROW];    // per-wave sigmoid(S) scratch [i][j]

    const int tid  = threadIdx.x;
    const int wave = tid >> 5;
    const int lane = tid & 31;
    const int b    = blockIdx.y;
    const int i0   = blockIdx.x * 128 + wave * 16;
    const int m    = lane & 15;         // lane-in-half
    const int h    = lane >> 4;         // half select

    const float* kbase = kM + (size_t)b * CATT * NPIX;
    const float* vbase = vM + (size_t)b * CATT * NPIX;

    // Kick off DMA for tile 0 into buffer 0 (one wave drives the TDM).
    if (wave == 0) {
        tdm_load_tile(lds_off(&sK[0][0]), kbase);
        tdm_load_tile(lds_off(&sV[0][0]), vbase);
    }

    // Preload Q^T fragments for this wave's 16 rows: 32 A-frags (K-steps of 4 ch).
    v2f qa[32];
    const float* qbase = qT + (((size_t)b * NPIX) + i0 + m) * CATT;
    #pragma unroll
    for (int s = 0; s < 32; ++s) {
        int c = 4 * s + 2 * h;                      // even -> 8B aligned
        qa[s] = *(const v2f*)(qbase + c);
    }

    v8f acc[8] = {};                                // O[16 i x 128 c]
    float* sSw = sS + wave * (16 * LROW);

    for (int j0 = 0; j0 < NPIX; j0 += 16) {
        int buf = (j0 >> 4) & 1;
        // All waves finished iteration j0-16 -> back buffer is free.
        __syncthreads();
        if (wave == 0) {
            if (j0 + 16 < NPIX) {
                tdm_load_tile(lds_off(&sK[buf ^ 1][0]), kbase + j0 + 16);
                tdm_load_tile(lds_off(&sV[buf ^ 1][0]), vbase + j0 + 16);
                // TDM completes in-order per wave: <=2 outstanding means the
                // current tile's two DMAs have landed.
                __builtin_amdgcn_s_wait_tensorcnt(2);
            } else {
                __builtin_amdgcn_s_wait_tensorcnt(0);
            }
        }
        __syncthreads();                            // current tile visible
        const float* sKb = &sK[buf][0];
        const float* sVb = &sV[buf][0];

        // ---- GEMM 1: S = Q^T * K (16x16 tile, K-dim 128 in steps of 4) ----
        v8f sfrag = {};
        #pragma unroll
        for (int s = 0; s < 32; ++s) {
            int c0 = 4 * s + 2 * h;
            v2f bk;
            bk.x = sKb[c0 * LROW + m];
            bk.y = sKb[(c0 + 1) * LROW + m];
            sfrag = __builtin_amdgcn_wmma_f32_16x16x4_f32(
                false, qa[s], false, bk, (short)0, sfrag, false, false);
        }

        // ---- sigmoid + per-wave transpose through LDS (DS pipe in-order) ----
        #pragma unroll
        for (int r = 0; r < 8; ++r) {
            float e  = sfrag[r];
            float sg = 1.0f / (1.0f + __expf(-e));
            sSw[(r + 8 * h) * LROW + m] = sg;       // [i_local][j_local]
        }
        asm volatile("" ::: "memory");              // keep store->load order

        // ---- GEMM 2: O[i,c] += P[i,j] * V^T[j,c] ----
        #pragma unroll
        for (int s = 0; s < 4; ++s) {
            v2f ap;                                 // A = P (16 i x 4 j)
            ap.x = sSw[m * LROW + 4 * s + 2 * h];
            ap.y = sSw[m * LROW + 4 * s + 2 * h + 1];
            #pragma unroll
            for (int ct = 0; ct < 8; ++ct) {
                int cc = ct * 16 + m;               // B = V^T (4 j x 16 c)
                v2f bv;
                bv.x = sVb[cc * LROW + 4 * s + 2 * h];
                bv.y = sVb[cc * LROW + 4 * s + 2 * h + 1];
                acc[ct] = __builtin_amdgcn_wmma_f32_16x16x4_f32(
                    false, ap, false, bv, (short)0, acc[ct], false, false);
            }
        }
    }

    // Store O in [b][i][c] layout (16 contiguous c per half-wave -> 64B stores).
    #pragma unroll
    for (int ct = 0; ct < 8; ++ct) {
        #pragma unroll
        for (int r = 0; r < 8; ++r) {
            int i = i0 + r + 8 * h;
            int c = ct * 16 + m;
            attn[(((size_t)b * NPIX) + i) * CATT + c] = acc[ct][r];
        }
    }
}

// ---------------------------------------------------------------------------
// Kernel 6: epilogue — o_v = prelu(bn(bt_w @ (attn + v*p) + bt_b))
// ---------------------------------------------------------------------------
__global__ __launch_bounds__(256)
void k_epilogue(const float* __restrict__ attn, const float* __restrict__ vM,
                const float* __restrict__ p,
                const float* __restrict__ bt_w, const float* __restrict__ bt_b,
                const float* __restrict__ g, const float* __restrict__ bb,
                const float* __restrict__ mm, const float* __restrict__ vv,
                const float* __restrict__ prelu2, float* __restrict__ out)
{
    int idx = blockIdx.x * 256 + threadIdx.x;       // 4*64*4096
    int i  = idx & 4095;
    int oc = (idx >> 12) & 63;
    int b  = idx >> 18;

    float pv = p[((size_t)b << 12) + i];
    float sum = bt_b[oc];
    const float* ar = attn + (((size_t)b * NPIX) + i) * CATT;
    const float* vb = vM + ((size_t)b * CATT) * NPIX + i;
    const float* wp = bt_w + oc * CATT;
    #pragma unroll 4
    for (int c = 0; c < CATT; ++c) {
        float pre = ar[c] + vb[(size_t)c * NPIX] * pv;
        sum += wp[c] * pre;
    }
    float scale = g[oc] * rsqrtf(vv[oc] + EPSV);
    float y = (sum - mm[oc]) * scale + bb[oc];
    float a = prelu2[0];
    out[idx] = (y >= 0.0f) ? y : a * y;
}

// ---------------------------------------------------------------------------
extern "C" void kernel_launch(void* const* d_in, const int* in_sizes, int n_in,
                              void* d_out, int out_size, void* d_ws, size_t ws_size,
                              hipStream_t stream)
{
    const float* em     = (const float*)d_in[0];
    const float* dm     = (const float*)d_in[1];
    const float* p      = (const float*)d_in[2];
    const float* en_w   = (const float*)d_in[3];
    const float* en_b   = (const float*)d_in[4];
    const float* bn1_g  = (const float*)d_in[5];
    const float* bn1_b  = (const float*)d_in[6];
    const float* bn1_m  = (const float*)d_in[7];
    const float* bn1_v  = (const float*)d_in[8];
    const float* prelu1 = (const float*)d_in[9];
    const float* dw_w   = (const float*)d_in[10];
    const float* dw_b   = (const float*)d_in[11];
    const float* de_w   = (const float*)d_in[12];
    const float* de_b   = (const float*)d_in[13];
    const float* q_w    = (const float*)d_in[14];
    const float* q_b    = (const float*)d_in[15];
    const float* k_w    = (const float*)d_in[16];
    const float* k_b    = (const float*)d_in[17];
    const float* v_w    = (const float*)d_in[18];
    const float* v_b    = (const float*)d_in[19];
    const float* bt_w   = (const float*)d_in[20];
    const float* bt_b   = (const float*)d_in[21];
    const float* bn2_g  = (const float*)d_in[22];
    const float* bn2_b  = (const float*)d_in[23];
    const float* bn2_m  = (const float*)d_in[24];
    const float* bn2_v  = (const float*)d_in[25];
    const float* prelu2 = (const float*)d_in[26];
    float* out = (float*)d_out;

    // Workspace layout (floats): ~39 MB total.
    float* ws       = (float*)d_ws;
    float* conv_out = ws;                       // 4*16*128*128 = 1,048,576
    float* f        = conv_out + 1048576;       // 4*48*4096    =   786,432
    float* qT       = f + 786432;               // 4*4096*128   = 2,097,152
    float* kM       = qT + 2097152;             // 2,097,152
    float* vM       = kM + 2097152;             // 2,097,152
    float* attn     = vM + 2097152;             // 2,097,152

    k_conv3x3<<<4096, 256, 0, stream>>>(em, en_w, en_b, bn1_g, bn1_b,
                                        bn1_m, bn1_v, prelu1, conv_out);
    k_pool_dw<<<1024, 256, 0, stream>>>(conv_out, dw_w, dw_b, f);
    k_de<<<2048, 256, 0, stream>>>(dm, de_w, de_b, f);
    k_qkv<<<8192, 256, 0, stream>>>(f, q_w, q_b, k_w, k_b, v_w, v_b, qT, kM, vM);

    dim3 agrid(NPIX / 128, BATCH);
    k_attention<<<agrid, 256, 0, stream>>>(qT, kM, vM, attn);

    k_epilogue<<<4096, 256, 0, stream>>>(attn, vM, p, bt_w, bt_b, bn2_g, bn2_b,
                                         bn2_m, bn2_v, prelu2, out);
}